// Net_36962488550095
// MI455X (gfx1250) — compile-verified
//
#include <hip/hip_runtime.h>
#include <math.h>

typedef __attribute__((ext_vector_type(2))) float v2f;
typedef __attribute__((ext_vector_type(8))) float v8f;

static constexpr int HWSZ = 1024 * 1024;   // full-res pixels per batch image

__device__ __forceinline__ int iclamp(int v, int lo, int hi) {
  return v < lo ? lo : (v > hi ? hi : v);
}

// ---------------------------------------------------------------------------
// Weight transpose: OIHW / [N,K] row-major  ->  B[k][n], K padded to Kp with 0
// ---------------------------------------------------------------------------
__global__ void wt_kernel(const float* __restrict__ w, float* __restrict__ wt,
                          int K, int Kp, int N) {
  int i = blockIdx.x * 256 + threadIdx.x;
  if (i >= Kp * N) return;
  int k = i / N, n = i % N;
  wt[i] = (k < K) ? w[n * K + k] : 0.f;
}

// ---------------------------------------------------------------------------
// im2col for 3x3 pad-1 conv. Output col[M][Kp], M = 2*OH*OW (pixel-major
// b,oy,ox), k = ci*9 + r*3 + c (matches OIHW weight order). nchw=1 -> input is
// NCHW (the lr tensor), nchw=0 -> input is pixel-major NHWC (our activations).
// ---------------------------------------------------------------------------
__global__ void im2col_kernel(const float* __restrict__ in, float* __restrict__ col,
                              int Cin, int IH, int IW, int OH, int OW,
                              int stride, int Kp, int nchw) {
  int M = 2 * OH * OW;
  int idx = blockIdx.x * 256 + threadIdx.x;
  if (idx >= M * Kp) return;
  int m = idx / Kp, k = idx % Kp;
  int K = Cin * 9;
  float v = 0.f;
  if (k < K) {
    int b = m / (OH * OW);
    int rem = m % (OH * OW);
    int oy = rem / OW, ox = rem % OW;
    int ci = k / 9;
    int rc = k % 9;
    int r = rc / 3, c = rc % 3;
    int iy = oy * stride - 1 + r;
    int ix = ox * stride - 1 + c;
    if (iy >= 0 && iy < IH && ix >= 0 && ix < IW) {
      v = nchw ? in[((b * Cin + ci) * IH + iy) * IW + ix]
               : in[((b * IH + iy) * IW + ix) * Cin + ci];
    }
  }
  col[idx] = v;
}

// ---------------------------------------------------------------------------
// f32 WMMA GEMM: C[M,N] = A[M,Kp] * B[Kp,N] + bias, optional ReLU.
// 128 threads = 4 waves per block; each wave owns one 16x16 output tile
// (4 consecutive M-tiles per block). K stepped by 4 via V_WMMA_F32_16X16X4_F32.
// Out-of-range rows/cols handled branchlessly: clamp address, zero by mask —
// the inner loop has no EXEC manipulation, and the wave reaches every WMMA
// with EXEC all-1s as the ISA requires. Kp must be a multiple of 4, and A rows
// must be 8-byte aligned (lda % 2 == 0, base 8B-aligned) for the b64 A loads.
// ---------------------------------------------------------------------------
__global__ void wmma_gemm_kernel(const float* __restrict__ A,
                                 const float* __restrict__ Bm,
                                 const float* __restrict__ bias,
                                 float* __restrict__ C,
                                 int M, int N, int Kp, int lda, int act) {
  int lane = threadIdx.x & 31;
  int wave = threadIdx.x >> 5;
  int tm = (blockIdx.x * 4 + wave) * 16;
  int tn = blockIdx.y * 16;

  int mA = tm + (lane & 15);           // A-frag row (lane&15)
  int nB = tn + (lane & 15);           // B-frag col (lane&15)
  int khalf = (lane >> 4) * 2;         // lanes 16-31 hold K+2,K+3

  // branchless bounds handling: clamp the address, zero the value
  int mCl = (mA < M) ? mA : 0;
  int nCl = (nB < N) ? nB : 0;
  float amask = (mA < M) ? 1.f : 0.f;
  float bmask = (nB < N) ? 1.f : 0.f;

  const float* Ap = A + (size_t)mCl * lda + khalf;   // 8B-aligned (khalf even)
  const float* Bp = Bm + (size_t)khalf * N + nCl;

  v8f acc = {};
  for (int k0 = 0; k0 < Kp; k0 += 4) {
    __builtin_prefetch(Ap + k0 + 32, 0, 1);          // global_prefetch_b8
    v2f a = *(const v2f*)(Ap + k0);                  // global_load_b64
    v2f b;
    b[0] = Bp[(size_t)k0 * N];
    b[1] = Bp[(size_t)(k0 + 1) * N];
    a *= amask;
    b *= bmask;
    acc = __builtin_amdgcn_wmma_f32_16x16x4_f32(
        /*neg_a=*/false, a, /*neg_b=*/false, b,
        /*c_mod=*/(short)0, acc, /*reuse_a=*/false, /*reuse_b=*/false);
  }

  // C/D layout: VGPR r -> m = tm + 8*(lane>>4) + r, n = tn + (lane&15)
  int mbase = tm + (lane >> 4) * 8;
  int n = tn + (lane & 15);
  if (n < N) {
    float bv = bias ? bias[n] : 0.f;
#pragma unroll
    for (int r = 0; r < 8; ++r) {
      int m = mbase + r;
      if (m < M) {
        float v = acc[r] + bv;
        if (act) v = fmaxf(v, 0.f);
        C[m * N + n] = v;
      }
    }
  }
}

// Reorder global head [2,4,4,64] pixel-major -> NCHW-flatten [2,1024]
__global__ void gflat_kernel(const float* __restrict__ g1, float* __restrict__ out) {
  int i = blockIdx.x * 256 + threadIdx.x;
  if (i >= 2 * 1024) return;
  int b = i / 1024, k = i % 1024;
  int c = k / 16, y = (k % 16) / 4, x = k % 4;
  out[i] = g1[((b * 4 + y) * 4 + x) * 64 + c];
}

// Fusion: fus[m][c] = relu(loc1[m][c] + g[b][c]), 512 pixels x 64 ch
__global__ void fuse_kernel(const float* __restrict__ loc1,
                            const float* __restrict__ gvec,
                            float* __restrict__ fus) {
  int i = blockIdx.x * 256 + threadIdx.x;
  if (i >= 512 * 64) return;
  int m = i / 64, c = i % 64;
  int b = m >> 8;                       // 256 pixels per batch
  fus[i] = fmaxf(loc1[i] + gvec[b * 64 + c], 0.f);
}

// Guide: per full-res pixel, 1x1 conv 3->16 relu, 1x1 16->1 sigmoid
__global__ void guide_kernel(const float* __restrict__ fr,
                             const float* __restrict__ w0, const float* __restrict__ b0,
                             const float* __restrict__ w1, const float* __restrict__ b1,
                             float* __restrict__ gd) {
  int p = blockIdx.x * 256 + threadIdx.x;
  if (p >= 2 * HWSZ) return;
  int b = p / HWSZ, rem = p % HWSZ;
  const float* base = fr + (size_t)b * 3 * HWSZ + rem;
  float r = base[0], g = base[HWSZ], bl = base[2 * HWSZ];
  float s = b1[0];
#pragma unroll
  for (int j = 0; j < 16; ++j) {
    float h = fmaf(w0[j * 3 + 0], r,
              fmaf(w0[j * 3 + 1], g,
              fmaf(w0[j * 3 + 2], bl, b0[j])));
    s = fmaf(w1[j], fmaxf(h, 0.f), s);
  }
  gd[p] = 1.f / (1.f + __expf(-s));
}

// Slice: trilinear-sample 12-channel affine from grid, apply to fr
// bg is pixel-major [ (b*16+gy)*16+gx ][ 96 ], grid[b,c,d,gy,gx] = bg[..][c*8+d]
__global__ void slice_kernel(const float* __restrict__ bg,
                             const float* __restrict__ gd,
                             const float* __restrict__ fr,
                             float* __restrict__ fin) {
  int p = blockIdx.x * 256 + threadIdx.x;
  if (p >= 2 * HWSZ) return;
  int b = p / HWSZ, rem = p % HWSZ;
  int y = rem >> 10, x = rem & 1023;

  float xs = (x + 0.5f) * (16.f / 1024.f) - 0.5f;
  float ys = (y + 0.5f) * (16.f / 1024.f) - 0.5f;
  float fx = floorf(xs), fy = floorf(ys);
  int x0 = iclamp((int)fx, 0, 15), x1 = x0 + 1 > 15 ? 15 : x0 + 1;
  int y0 = iclamp((int)fy, 0, 15), y1 = y0 + 1 > 15 ? 15 : y0 + 1;
  float wx = xs - fx, wy = ys - fy;

  float gz = gd[p] * 8.f - 0.5f;
  float fz = floorf(gz);
  int z0 = iclamp((int)fz, 0, 7), z1 = z0 + 1 > 7 ? 7 : z0 + 1;
  float wz = gz - fz;

  int zi[2] = {z0, z1};   float wzf[2] = {1.f - wz, wz};
  int yi[2] = {y0, y1};   float wyf[2] = {1.f - wy, wy};
  int xi[2] = {x0, x1};   float wxf[2] = {1.f - wx, wx};

  float acc[12];
#pragma unroll
  for (int c = 0; c < 12; ++c) acc[c] = 0.f;

  for (int a = 0; a < 2; ++a)
    for (int yy = 0; yy < 2; ++yy)
      for (int xx = 0; xx < 2; ++xx) {
        float w = wzf[a] * wyf[yy] * wxf[xx];
        const float* g = bg + ((b * 16 + yi[yy]) * 16 + xi[xx]) * 96;
#pragma unroll
        for (int c = 0; c < 12; ++c) acc[c] += w * g[c * 8 + zi[a]];
      }

  const float* base = fr + (size_t)b * 3 * HWSZ + rem;
  float r = base[0], g = base[HWSZ], bl = base[2 * HWSZ];
  float* ob = fin + (size_t)b * 3 * HWSZ + rem;
#pragma unroll
  for (int o = 0; o < 3; ++o)
    ob[o * HWSZ] = fmaf(acc[o * 4 + 0], r,
                   fmaf(acc[o * 4 + 1], g,
                   fmaf(acc[o * 4 + 2], bl, acc[o * 4 + 3])));
}

// ---------------------------------------------------------------------------
extern "C" void kernel_launch(void* const* d_in, const int* in_sizes, int n_in,
                              void* d_out, int out_size, void* d_ws, size_t ws_size,
                              hipStream_t stream) {
  (void)in_sizes; (void)n_in; (void)out_size; (void)ws_size;

  const float* lr   = (const float*)d_in[0];
  const float* fr   = (const float*)d_in[1];
  const float* sw0  = (const float*)d_in[2];  const float* sb0  = (const float*)d_in[3];
  const float* sw1  = (const float*)d_in[4];  const float* sb1  = (const float*)d_in[5];
  const float* sw2  = (const float*)d_in[6];  const float* sb2  = (const float*)d_in[7];
  const float* sw3  = (const float*)d_in[8];  const float* sb3  = (const float*)d_in[9];
  const float* lw0  = (const float*)d_in[10]; const float* lb0  = (const float*)d_in[11];
  const float* lw1  = (const float*)d_in[12]; const float* lb1  = (const float*)d_in[13];
  const float* gcw0 = (const float*)d_in[14]; const float* gcb0 = (const float*)d_in[15];
  const float* gcw1 = (const float*)d_in[16]; const float* gcb1 = (const float*)d_in[17];
  const float* fcw0 = (const float*)d_in[18]; const float* fcb0 = (const float*)d_in[19];
  const float* fcw1 = (const float*)d_in[20]; const float* fcb1 = (const float*)d_in[21];
  const float* fcw2 = (const float*)d_in[22]; const float* fcb2 = (const float*)d_in[23];
  const float* bgw  = (const float*)d_in[24]; const float* bgb  = (const float*)d_in[25];
  const float* guw0 = (const float*)d_in[26]; const float* gub0 = (const float*)d_in[27];
  const float* guw1 = (const float*)d_in[28]; const float* gub1 = (const float*)d_in[29];

  float* out = (float*)d_out;
  float* gd  = out;                  // [2,1024,1024]
  float* fin = out + 2 * HWSZ;       // [2,3,1024,1024]

  // workspace carve-up (floats)
  float* W = (float*)d_ws;
  size_t off = 0;
  auto alloc = [&](size_t n) { float* p = W + off; off += n; return p; };
  float* col  = alloc(32768 * 28);   // im2col scratch (max over all convs)
  float* wt   = alloc(1024 * 256);   // transposed-weight scratch (max)
  float* x0   = alloc(32768 * 8);    // [2,128,128,8] pixel-major
  float* x1   = alloc(8192 * 16);
  float* x2   = alloc(2048 * 32);
  float* x3   = alloc(512 * 64);
  float* loc0 = alloc(512 * 64);
  float* loc1 = alloc(512 * 64);
  float* g0   = alloc(128 * 64);
  float* g1   = alloc(32 * 64);
  float* gfl  = alloc(2 * 1024);
  float* fc0o = alloc(2 * 256);
  float* fc1o = alloc(2 * 128);
  float* fc2o = alloc(2 * 64);
  float* fus  = alloc(512 * 64);
  float* bgb_ = alloc(512 * 96);     // bilateral grid, pixel-major x 96ch

  auto cdiv = [](int a, int b) { return (a + b - 1) / b; };

  auto gemm = [&](const float* A, int lda, const float* Bm, const float* bias,
                  float* C, int M, int N, int Kp, int act) {
    dim3 g(cdiv(M, 64), cdiv(N, 16));   // 4 waves/block, one M-tile each
    wmma_gemm_kernel<<<g, 128, 0, stream>>>(A, Bm, bias, C, M, N, Kp, lda, act);
  };

  auto conv3x3 = [&](const float* in, int nchw, int Cin, int IH, int stride,
                     const float* w, const float* bias, int Cout, float* o, int act) {
    int OH = (IH - 1) / stride + 1;          // pad=1, k=3
    int K = Cin * 9, Kp = (K + 3) & ~3;
    int M = 2 * OH * OH;
    wt_kernel<<<cdiv(Kp * Cout, 256), 256, 0, stream>>>(w, wt, K, Kp, Cout);
    im2col_kernel<<<cdiv(M * Kp, 256), 256, 0, stream>>>(in, col, Cin, IH, IH,
                                                         OH, OH, stride, Kp, nchw);
    gemm(col, Kp, wt, bias, o, M, Cout, Kp, act);
  };

  // splat conv stack: 256 -> 128 -> 64 -> 32 -> 16
  conv3x3(lr, 1, 3, 256, 2, sw0, sb0, 8, x0, 1);
  conv3x3(x0, 0, 8, 128, 2, sw1, sb1, 16, x1, 1);
  conv3x3(x1, 0, 16, 64, 2, sw2, sb2, 32, x2, 1);
  conv3x3(x2, 0, 32, 32, 2, sw3, sb3, 64, x3, 1);
  // local head
  conv3x3(x3, 0, 64, 16, 1, lw0, lb0, 64, loc0, 1);
  conv3x3(loc0, 0, 64, 16, 1, lw1, lb1, 64, loc1, 0);
  // global head convs: 16 -> 8 -> 4
  conv3x3(x3, 0, 64, 16, 2, gcw0, gcb0, 64, g0, 1);
  conv3x3(g0, 0, 64, 8, 2, gcw1, gcb1, 64, g1, 1);
  // flatten (NCHW order) + FC stack
  gflat_kernel<<<cdiv(2 * 1024, 256), 256, 0, stream>>>(g1, gfl);
  wt_kernel<<<cdiv(1024 * 256, 256), 256, 0, stream>>>(fcw0, wt, 1024, 1024, 256);
  gemm(gfl, 1024, wt, fcb0, fc0o, 2, 256, 1024, 1);
  wt_kernel<<<cdiv(256 * 128, 256), 256, 0, stream>>>(fcw1, wt, 256, 256, 128);
  gemm(fc0o, 256, wt, fcb1, fc1o, 2, 128, 256, 1);
  wt_kernel<<<cdiv(128 * 64, 256), 256, 0, stream>>>(fcw2, wt, 128, 128, 64);
  gemm(fc1o, 128, wt, fcb2, fc2o, 2, 64, 128, 0);
  // fusion + 1x1 conv to 96-channel bilateral grid (im2col-free: 1x1, NHWC)
  fuse_kernel<<<cdiv(512 * 64, 256), 256, 0, stream>>>(loc1, fc2o, fus);
  wt_kernel<<<cdiv(64 * 96, 256), 256, 0, stream>>>(bgw, wt, 64, 64, 96);
  gemm(fus, 64, wt, bgb, bgb_, 512, 96, 64, 0);
  // full-res guide + slice/apply
  guide_kernel<<<cdiv(2 * HWSZ, 256), 256, 0, stream>>>(fr, guw0, gub0, guw1, gub1, gd);
  slice_kernel<<<cdiv(2 * HWSZ, 256), 256, 0, stream>>>(bgb_, gd, fr, fin);
}